// FastRCNNOutputLayers_23364622090718
// MI455X (gfx1250) — compile-verified
//
#include <hip/hip_runtime.h>
#include <stdint.h>

#define R_     10000
#define K_     80
#define NC     81
#define MPRE   2048
#define NIMG   8
#define NBINS  4608
#define GCAP   8192
#define CSTRIDE 12
#define TOPK_  100
#define OUTC   85   // 4 box + 1 score + 80 logits
#define CHUNK  64   // NMS rows per async-staged LDS chunk

__device__ __forceinline__ float clampf(float v, float lo, float hi) {
  return fminf(fmaxf(v, lo), hi);
}

__device__ __forceinline__ void async_lds_b128(unsigned ldsOff, const void* gsrc) {
  unsigned long long ga = (unsigned long long)(uintptr_t)gsrc;
  asm volatile("global_load_async_to_lds_b128 %0, %1, off" :: "v"(ldsOff), "v"(ga) : "memory");
}
__device__ __forceinline__ void async_lds_b64(unsigned ldsOff, const void* gsrc) {
  unsigned long long ga = (unsigned long long)(uintptr_t)gsrc;
  asm volatile("global_load_async_to_lds_b64 %0, %1, off" :: "v"(ldsOff), "v"(ga) : "memory");
}
__device__ __forceinline__ void wait_async0() {
  asm volatile("s_wait_asynccnt 0" ::: "memory");
}

// ---------------- zero scratch counters ----------------
__global__ void k_zero(unsigned* __restrict__ p, int n) {
  int i = blockIdx.x * blockDim.x + threadIdx.x;
  if (i < n) p[i] = 0u;
}

// ---------------- histogram of float-bits of scores > 0.05 ----------------
__global__ void k_hist(const float* __restrict__ scores, unsigned* __restrict__ hist) {
  int img = blockIdx.y;
  int e = blockIdx.x * 256 + threadIdx.x;
  if (e >= R_ * K_) return;
  int r = e / K_, c = e - r * K_;
  const float* sp = scores + (size_t)img * R_ * NC + (size_t)r * NC + c;
  __builtin_prefetch(sp + NC * 64, 0, 1);   // global_prefetch_b8
  float s = *sp;
  if (s > 0.05f) {
    unsigned bits = __float_as_uint(s);
    unsigned base = __float_as_uint(0.05f) + 1u;   // s > 0.05f  <=>  bits >= base
    unsigned b = (bits - base) >> 13;
    if (b >= NBINS) b = NBINS - 1;
    atomicAdd(&hist[img * NBINS + b], 1u);
  }
}

// ---------------- find bin threshold where cumulative-from-top >= 2048 ----------------
__global__ void k_scan(const unsigned* __restrict__ hist, unsigned* __restrict__ gthresh) {
  int img = blockIdx.x;
  if (threadIdx.x != 0) return;
  unsigned cum = 0; int B = 0;
  for (int b = NBINS - 1; b >= 0; --b) {
    cum += hist[img * NBINS + b];
    if (cum >= MPRE) { B = b; break; }
  }
  gthresh[img] = (__float_as_uint(0.05f) + 1u) + ((unsigned)B << 13);
}

// ---------------- gather all candidates above bin threshold ----------------
__global__ void k_gather(const float* __restrict__ scores, const unsigned* __restrict__ gthresh,
                         unsigned* __restrict__ gcount, unsigned long long* __restrict__ gbuf) {
  int img = blockIdx.y;
  int e = blockIdx.x * 256 + threadIdx.x;
  if (e >= R_ * K_) return;
  int r = e / K_, c = e - r * K_;
  const float* sp = scores + (size_t)img * R_ * NC + (size_t)r * NC + c;
  __builtin_prefetch(sp + NC * 64, 0, 1);
  float s = *sp;
  if (s > 0.05f) {
    unsigned bits = __float_as_uint(s);
    if (bits >= gthresh[img]) {
      unsigned pos = atomicAdd(&gcount[img], 1u);
      if (pos < GCAP) {
        unsigned idx = (unsigned)(r * K_ + c);
        gbuf[(size_t)img * GCAP + pos] =
            ((unsigned long long)bits << 32) | (unsigned long long)(0xFFFFFFFFu - idx);
      }
    }
  }
}

// ---------------- bitonic sort (descending) of up to 8192 keys in LDS ----------------
__global__ __launch_bounds__(1024) void k_sort(const unsigned* __restrict__ gcount,
                                               const unsigned long long* __restrict__ gbuf,
                                               unsigned long long* __restrict__ candKeys) {
  __shared__ __align__(16) unsigned long long keys[GCAP];   // 64 KB LDS
  int img = blockIdx.x;
  unsigned C = gcount[img]; if (C > GCAP) C = GCAP;
  const unsigned long long* src = gbuf + (size_t)img * GCAP;

  for (int i = threadIdx.x; i < GCAP; i += 1024) {
    if ((unsigned)i < C) {
      async_lds_b64((unsigned)(uintptr_t)&keys[i], src + i);
    } else {
      keys[i] = 0ull;   // sentinel -> sorts last, marks invalid candidate
    }
  }
  wait_async0();
  __syncthreads();

  for (int k = 2; k <= GCAP; k <<= 1) {
    for (int j = k >> 1; j > 0; j >>= 1) {
      for (int i = threadIdx.x; i < GCAP; i += 1024) {
        int ixj = i ^ j;
        if (ixj > i) {
          unsigned long long a = keys[i], b = keys[ixj];
          bool sw = ((i & k) == 0) ? (a < b) : (a > b);   // descending
          if (sw) { keys[i] = b; keys[ixj] = a; }
        }
      }
      __syncthreads();
    }
  }
  for (int i = threadIdx.x; i < MPRE; i += 1024)
    candKeys[(size_t)img * MPRE + i] = keys[i];
}

// ---------------- decode candidates: clip boxes, class-offset, validity bitset ----------------
__global__ void k_cand(const float* __restrict__ boxes, const unsigned long long* __restrict__ candKeys,
                       float* __restrict__ candF, unsigned* __restrict__ validWords) {
  int img = blockIdx.y;
  int slot = blockIdx.x * 256 + threadIdx.x;
  unsigned long long key = candKeys[(size_t)img * MPRE + slot];
  int valid = (key != 0ull) ? 1 : 0;
  unsigned idx = valid ? (0xFFFFFFFFu - (unsigned)(key & 0xFFFFFFFFull)) : 0u;
  int r = (int)(idx / K_), c = (int)(idx - (unsigned)r * K_);
  const float* bp = boxes + (size_t)img * R_ * (K_ * 4) + (size_t)r * (K_ * 4) + c * 4;
  float x1 = clampf(bp[0], 0.f, 1333.f);
  float y1 = clampf(bp[1], 0.f, 800.f);
  float x2 = clampf(bp[2], 0.f, 1333.f);
  float y2 = clampf(bp[3], 0.f, 800.f);
  float off = (float)c * 4096.0f;
  float cs = __uint_as_float((unsigned)(key >> 32));
  float* o = candF + ((size_t)img * MPRE + slot) * CSTRIDE;
  o[0] = x1 + off; o[1] = y1 + off; o[2] = x2 + off; o[3] = y2 + off;   // offset boxes (NMS)
  o[4] = x1; o[5] = y1; o[6] = x2; o[7] = y2;                           // clipped boxes (output)
  o[8] = cs;
  ((int*)o)[9]  = r;
  ((int*)o)[10] = valid;
  o[11] = 0.f;
  unsigned m = __builtin_amdgcn_ballot_w32(valid != 0);
  if ((threadIdx.x & 31) == 0)
    validWords[img * 64 + slot / 32] = m;
}

// ---------------- suppression bitmask: 32 rows/block, box table async-staged in LDS ----------------
__global__ __launch_bounds__(256) void k_mask(const float* __restrict__ candF,
                                              unsigned* __restrict__ mask) {
  __shared__ __align__(16) float4 sob[MPRE];   // 32 KB: all 2048 offset boxes
  int img = blockIdx.y;
  int i0 = blockIdx.x * 32;
  const float* cbase = candF + (size_t)img * MPRE * CSTRIDE;

  // async global->LDS stage of the full candidate box table (16 B per slot)
  for (int s = threadIdx.x; s < MPRE; s += 256)
    async_lds_b128((unsigned)(uintptr_t)&sob[s], cbase + (size_t)s * CSTRIDE);
  wait_async0();
  __syncthreads();

  int lane = threadIdx.x & 31, wave = threadIdx.x >> 5;
  for (int rr = wave; rr < 32; rr += 8) {   // 4 rows per wave
    int i = i0 + rr;
    float4 bi = sob[i];
    float ai = fmaxf(bi.z - bi.x, 0.f) * fmaxf(bi.w - bi.y, 0.f);
    for (int w = 0; w < 64; ++w) {
      int j = w * 32 + lane;
      float4 bj = sob[j];
      float aj = fmaxf(bj.z - bj.x, 0.f) * fmaxf(bj.w - bj.y, 0.f);
      float lx = fmaxf(bi.x, bj.x), ly = fmaxf(bi.y, bj.y);
      float rx = fminf(bi.z, bj.z), ry = fminf(bi.w, bj.w);
      float iw = fmaxf(rx - lx, 0.f), ih = fmaxf(ry - ly, 0.f);
      float inter = iw * ih;
      float iou = inter / (ai + aj - inter + 1e-9f);
      bool sup = (iou > 0.5f) && (j > i);
      unsigned m = __builtin_amdgcn_ballot_w32(sup);
      if (lane == 0) mask[((size_t)img * MPRE + i) * 64 + w] = m;
    }
  }
}

// ---------------- greedy NMS: one wave32/image, double-buffered async row pipeline ----------------
__global__ __launch_bounds__(32) void k_nms(const unsigned* __restrict__ validWords,
                                            const unsigned* __restrict__ mask,
                                            unsigned* __restrict__ keepWords) {
  __shared__ __align__(16) unsigned buf[2][CHUNK * 64];   // 2 x 16 KB
  int img = blockIdx.x;
  int lane = threadIdx.x;          // blockDim == 32 (one wave)
  unsigned vw0 = validWords[img * 64 + lane];
  unsigned vw1 = validWords[img * 64 + lane + 32];
  unsigned sup0 = 0u, sup1 = 0u;
  const unsigned* mbase = mask + (size_t)img * MPRE * 64;

  // issue async copy of 64 rows (16 KB contiguous) into buf[b]
  auto issue = [&](int chunk, int b) {
    const unsigned* src = mbase + (size_t)chunk * CHUNK * 64;
    for (int t = 0; t < 32; ++t) {                 // 1024 x 16B, 32 per lane
      int q = t * 32 + lane;                        // 16-byte quad index
      async_lds_b128((unsigned)(uintptr_t)&buf[b][q * 4], src + (size_t)q * 4);
    }
  };

  issue(0, 0);
  for (int chunk = 0; chunk < MPRE / CHUNK; ++chunk) {
    int cur = chunk & 1;
    wait_async0();                                  // drain this chunk's DMA
    __builtin_amdgcn_wave_barrier();
    if (chunk + 1 < MPRE / CHUNK) issue(chunk + 1, cur ^ 1);   // overlap next chunk
    for (int ii = 0; ii < CHUNK; ++ii) {
      int i = chunk * CHUNK + ii;
      int iw = i >> 5;                              // uniform
      int owner = iw & 31;
      unsigned sw = __shfl((iw < 32) ? sup0 : sup1, owner, 32);
      unsigned vv = __shfl((iw < 32) ? vw0  : vw1,  owner, 32);
      unsigned bit = 1u << (i & 31);
      bool act = (vv & bit) && !(sw & bit);
      if (act) {
        sup0 |= buf[cur][ii * 64 + lane];
        sup1 |= buf[cur][ii * 64 + lane + 32];
      }
    }
  }
  keepWords[img * 64 + lane]      = vw0 & ~sup0;
  keepWords[img * 64 + lane + 32] = vw1 & ~sup1;
}

// ---------------- compact first 100 kept, emit [box, score, logits-row], zero-fill rest ----------------
__global__ void k_out(const float* __restrict__ candF, const unsigned* __restrict__ keepWords,
                      const float* __restrict__ logits, float* __restrict__ out) {
  __shared__ int sSel[TOPK_], sR[TOPK_];
  __shared__ int sK;
  int img = blockIdx.x;
  if (threadIdx.x == 0) {
    int n = 0;
    for (int w = 0; w < 64 && n < TOPK_; ++w) {
      unsigned bitsw = keepWords[img * 64 + w];
      while (bitsw && n < TOPK_) {
        int b = __ffs(bitsw) - 1;
        bitsw &= bitsw - 1u;
        int slot = w * 32 + b;
        sSel[n] = slot;
        sR[n] = ((const int*)candF)[((size_t)img * MPRE + slot) * CSTRIDE + 9];
        ++n;
      }
    }
    sK = n;
  }
  __syncthreads();
  for (int e = threadIdx.x; e < TOPK_ * OUTC; e += blockDim.x) {
    int n = e / OUTC, col = e - n * OUTC;
    float v = 0.f;
    if (n < sK) {
      int slot = sSel[n];
      const float* cd = candF + ((size_t)img * MPRE + slot) * CSTRIDE;
      if (col < 4)       v = cd[4 + col];
      else if (col == 4) v = cd[8];
      else               v = logits[(size_t)img * R_ * NC + (size_t)sR[n] * NC + (col - 5)];
    }
    out[(size_t)img * TOPK_ * OUTC + e] = v;
  }
}

extern "C" void kernel_launch(void* const* d_in, const int* in_sizes, int n_in,
                              void* d_out, int out_size, void* d_ws, size_t ws_size,
                              hipStream_t stream) {
  const float* boxes  = (const float*)d_in[0];   // (8, 10000, 320)
  const float* scores = (const float*)d_in[1];   // (8, 10000, 81)
  const float* logits = (const float*)d_in[2];   // (8, 10000, 81)
  float* out = (float*)d_out;                    // (8, 100, 85)

  char* ws = (char*)d_ws;
  const size_t off_hist    = 0;                                        // 147456
  const size_t off_gcount  = off_hist    + (size_t)NIMG * NBINS * 4;
  const size_t off_gthresh = off_gcount  + NIMG * 4;
  const size_t off_gbuf    = off_gthresh + NIMG * 4;                   // 524288
  const size_t off_ckeys   = off_gbuf    + (size_t)NIMG * GCAP * 8;    // 131072
  const size_t off_cand    = off_ckeys   + (size_t)NIMG * MPRE * 8;    // 786432 (16B-aligned base)
  const size_t off_valid   = off_cand    + (size_t)NIMG * MPRE * CSTRIDE * 4;
  const size_t off_mask    = off_valid   + (size_t)NIMG * 64 * 4;      // 4 MB (16B-aligned)
  const size_t off_keep    = off_mask    + (size_t)NIMG * MPRE * 64 * 4;

  unsigned*            hist     = (unsigned*)(ws + off_hist);
  unsigned*            gcount   = (unsigned*)(ws + off_gcount);
  unsigned*            gthresh  = (unsigned*)(ws + off_gthresh);
  unsigned long long*  gbuf     = (unsigned long long*)(ws + off_gbuf);
  unsigned long long*  candKeys = (unsigned long long*)(ws + off_ckeys);
  float*               candF    = (float*)(ws + off_cand);
  unsigned*            validW   = (unsigned*)(ws + off_valid);
  unsigned*            maskW    = (unsigned*)(ws + off_mask);
  unsigned*            keepW    = (unsigned*)(ws + off_keep);

  int nzero = NIMG * NBINS + NIMG;           // hist + gcount (contiguous)
  k_zero<<<(nzero + 255) / 256, 256, 0, stream>>>(hist, nzero);

  dim3 gscan((R_ * K_ + 255) / 256, NIMG);
  k_hist  <<<gscan, 256, 0, stream>>>(scores, hist);
  k_scan  <<<NIMG,  32,  0, stream>>>(hist, gthresh);
  k_gather<<<gscan, 256, 0, stream>>>(scores, gthresh, gcount, gbuf);
  k_sort  <<<NIMG, 1024, 0, stream>>>(gcount, gbuf, candKeys);

  dim3 gcand(MPRE / 256, NIMG);
  k_cand<<<gcand, 256, 0, stream>>>(boxes, candKeys, candF, validW);

  dim3 gmask(MPRE / 32, NIMG);
  k_mask<<<gmask, 256, 0, stream>>>(candF, maskW);
  k_nms <<<NIMG, 32, 0, stream>>>(validW, maskW, keepW);
  k_out <<<NIMG, 256, 0, stream>>>(candF, keepW, logits, out);
}